// ForwardFunctionalMap_55637006352876
// MI455X (gfx1250) — compile-verified
//
#include <hip/hip_runtime.h>
#include <math.h>
#include <stdint.h>

// ---------------------------------------------------------------------------
// ForwardFunctionalMap for MI455X (gfx1250, wave32)
//   stage 1: A_x = pinv_x @ feat_x, A_y = pinv_y @ feat_y
//            f32 WMMA 16x16x4, feat staged via async global->LDS (ASYNCcnt)
//   stage 2: AA_xx = A_x A_x^T, AA_yx = A_y A_x^T  (f32 WMMA 16x16x4)
//   stage 3: per (b,i) SPD solve; matrix tile DMA'd into bank-padded LDS by
//            the Tensor Data Mover (TENSORcnt), Gaussian elim, no pivoting
// ---------------------------------------------------------------------------

typedef __attribute__((ext_vector_type(2))) float v2f;
typedef __attribute__((ext_vector_type(8))) float v8f;
typedef __attribute__((ext_vector_type(4))) unsigned int v4u;
typedef __attribute__((ext_vector_type(8))) int v8i;
typedef __attribute__((ext_vector_type(4))) int v4i;

#define LMBDA_ 100.0f
#define B_ 32
#define K_ 128
#define V_ 5000
#define C_ 256
#define CH_ 32                 // V-rows staged per chunk
#define NFULL_ (V_ / CH_)      // 156 full chunks
#define TAIL_ (V_ - NFULL_ * CH_)  // 8-row tail chunk (chunk index NFULL_)

// ---------------------------------------------------------------------------
// Kernel 0: per-batch scale[b] = max(max(evals_x[b,:]), max(evals_y[b,:]))
// ---------------------------------------------------------------------------
__global__ __launch_bounds__(128) void fm_scale_kernel(
    const float* __restrict__ evx, const float* __restrict__ evy,
    float* __restrict__ scale) {
  __shared__ float s[128];
  const int b = blockIdx.x;
  const int t = threadIdx.x;
  s[t] = fmaxf(evx[b * K_ + t], evy[b * K_ + t]);
  __syncthreads();
  for (int off = 64; off > 0; off >>= 1) {
    if (t < off) s[t] = fmaxf(s[t], s[t + off]);
    __syncthreads();
  }
  if (t == 0) scale[b] = s[0];
}

// ---------------------------------------------------------------------------
// Kernel 1: projection GEMM.
// Block = 4 waves sharing one 16-wide C tile, covering 64 K-rows. The shared
// B operand (feat rows) is double-buffered in LDS via
// global_load_async_to_lds_b128 (ASYNCcnt); every thread issues exactly one
// async b128 per chunk so s_wait_asynccnt + barrier form the handshake.
// 156 full 32-row chunks (compile-time 8-WMMA inner loop) + one peeled
// 8-row tail, so the steady state has no remainder loop.
// ---------------------------------------------------------------------------
__global__ __launch_bounds__(128) void fm_proj_kernel(
    const float* __restrict__ pinv_x, const float* __restrict__ feat_x,
    const float* __restrict__ pinv_y, const float* __restrict__ feat_y,
    float* __restrict__ Ax, float* __restrict__ Ay) {
  __shared__ float bsh[2][CH_ * 16];  // 2 x 2 KB double buffer

  const int lane = threadIdx.x;
  const int wy   = threadIdx.y;       // wave id 0..3
  const int half = lane >> 4;
  const int lm   = lane & 15;
  const int ct   = blockIdx.x * 16;
  const int kt   = blockIdx.y * 64 + wy * 16;
  const int b    = blockIdx.z >> 1;
  const bool isY = (blockIdx.z & 1) != 0;

  const float* pinv = (isY ? pinv_y : pinv_x) + (size_t)b * K_ * V_;
  const float* feat = (isY ? feat_y : feat_x) + (size_t)b * V_ * C_;
  float* out = (isY ? Ay : Ax) + (size_t)b * K_ * C_;

  const float* ap = pinv + (size_t)(kt + lm) * V_ + half * 2;

  // async-copy assignment: thread t moves 16B of feat row (rowi, 4 cols)
  const int t    = wy * 32 + lane;
  const int rowi = t >> 2;            // 0..31
  const int cseg = (t & 3) * 4;       // 0,4,8,12
  // low 32 bits of a generic pointer to LDS == byte offset within LDS
  const uint32_t lds0 = (uint32_t)(uintptr_t)(&bsh[0][0]);

  auto issue = [&](int c) {
    const int v0   = c * CH_;
    const int rows = (V_ - v0 < CH_) ? (V_ - v0) : CH_;
    const int gr   = v0 + ((rowi < rows) ? rowi : rows - 1);  // clamp, no branch
    const float* g = feat + (size_t)gr * C_ + (ct + cseg);
    const uint32_t l =
        lds0 + (uint32_t)((((c & 1) * (CH_ * 16)) + rowi * 16 + cseg) * 4);
    asm volatile("global_load_async_to_lds_b128 %0, %1, off"
                 :: "v"(l), "v"(g) : "memory");
  };

  v8f acc = {};
  issue(0);
  for (int c = 0; c < NFULL_; ++c) {
    issue(c + 1);                                       // c+1 <= NFULL_ (tail)
    asm volatile("s_wait_asynccnt 0x1" ::: "memory");   // chunk c landed
    __syncthreads();  // all waves' async writes visible to all consumers

    const int v0 = c * CH_;
    const float* bbuf = &bsh[c & 1][0];
    __builtin_prefetch(ap + v0 + 2 * CH_, 0, 3);        // near-cache A hint

#pragma unroll
    for (int kk = 0; kk < CH_; kk += 4) {               // constant trip: 8 WMMA
      v2f a = *reinterpret_cast<const v2f*>(ap + v0 + kk);
      v2f bb;
      bb.x = bbuf[(kk + 2 * half) * 16 + lm];           // conflict-free DS
      bb.y = bbuf[(kk + 2 * half + 1) * 16 + lm];
      acc = __builtin_amdgcn_wmma_f32_16x16x4_f32(
          false, a, false, bb, (short)0, acc, false, false);
    }
    __syncthreads();  // everyone done reading buf (c&1) before re-issue
  }

  // peeled 8-row tail (chunk NFULL_, buffer NFULL_&1 == 0)
  asm volatile("s_wait_asynccnt 0x0" ::: "memory");
  __syncthreads();
  {
    const int v0 = NFULL_ * CH_;
    const float* bbuf = &bsh[NFULL_ & 1][0];
#pragma unroll
    for (int kk = 0; kk < TAIL_; kk += 4) {
      v2f a = *reinterpret_cast<const v2f*>(ap + v0 + kk);
      v2f bb;
      bb.x = bbuf[(kk + 2 * half) * 16 + lm];
      bb.y = bbuf[(kk + 2 * half + 1) * 16 + lm];
      acc = __builtin_amdgcn_wmma_f32_16x16x4_f32(
          false, a, false, bb, (short)0, acc, false, false);
    }
  }

  // D layout: VGPR g holds M = g + 8*half; lane's N = lm
  float* o = out + (size_t)(kt + 8 * half) * C_ + (ct + lm);
#pragma unroll
  for (int g = 0; g < 8; ++g) o[(size_t)g * C_] = acc[g];
}

// ---------------------------------------------------------------------------
// Kernel 2: Gram GEMMs. out[k,l] = sum_c L[k,c] * A_x[l,c]; contraction along
// C so BOTH operands are contiguous float2 loads per lane (data is L2-hot).
// ---------------------------------------------------------------------------
__global__ __launch_bounds__(32) void fm_gram_kernel(
    const float* __restrict__ Ax, const float* __restrict__ Ay,
    float* __restrict__ AAxx, float* __restrict__ AAyx) {
  const int lane = threadIdx.x;
  const int half = lane >> 4;
  const int lm   = lane & 15;
  const int lt   = blockIdx.x * 16;   // l tile (columns)
  const int kt   = blockIdx.y * 16;   // k tile (rows)
  const int b    = blockIdx.z >> 1;
  const bool isYX = (blockIdx.z & 1) != 0;

  const float* L = (isYX ? Ay : Ax) + (size_t)b * K_ * C_;
  const float* R = Ax + (size_t)b * K_ * C_;
  float* out = (isYX ? AAyx : AAxx) + (size_t)b * K_ * K_;

  const float* ap = L + (size_t)(kt + lm) * C_ + half * 2;
  const float* bp = R + (size_t)(lt + lm) * C_ + half * 2;

  v8f acc = {};
#pragma unroll 8
  for (int c = 0; c < C_; c += 4) {
    v2f a  = *reinterpret_cast<const v2f*>(ap + c);
    v2f bb = *reinterpret_cast<const v2f*>(bp + c);
    acc = __builtin_amdgcn_wmma_f32_16x16x4_f32(
        false, a, false, bb, (short)0, acc, false, false);
  }

  float* o = out + (size_t)(kt + 8 * half) * K_ + (lt + lm);
#pragma unroll
  for (int g = 0; g < 8; ++g) o[(size_t)g * K_] = acc[g];
}

// ---------------------------------------------------------------------------
// Kernel 3: one block per (b, i): solve (AA_xx[b] + lambda*diag(MASK[b,i,:]))
//           x = AA_yx[b,i,:].  SPD -> Gaussian elimination without pivoting.
//
// The 128x128 f32 tile is DMA'd into LDS by the Tensor Data Mover with
// pad_enable=1, pad_interval=6 (every 128 DWORDs), pad_amount=0 (1 DWORD):
// hardware produces the 129-stride bank-padded layout directly. The pad
// column (untouched by the TDM) holds the RHS.
// ---------------------------------------------------------------------------
#define MS(r, c) Msh[(size_t)(r) * (K_ + 1) + (c)]
#define RHS(r)   Msh[(size_t)(r) * (K_ + 1) + K_]

__global__ __launch_bounds__(128) void fm_solve_kernel(
    const float* __restrict__ AAxx, const float* __restrict__ AAyx,
    const float* __restrict__ evx, const float* __restrict__ evy,
    const float* __restrict__ scale, float* __restrict__ out) {
  __shared__ float Msh[(size_t)K_ * (K_ + 1)];  // 66048 B (< 320 KB/WGP)

  const int i = blockIdx.x;   // mask row index
  const int b = blockIdx.y;
  const int r = threadIdx.x;  // this thread owns matrix row r

  // --- wave 0: TDM load of AA_xx[b] (64 KB) into padded LDS ---------------
  if (threadIdx.x < 32) {
    const uint64_t ga = (uint64_t)(uintptr_t)(AAxx + (size_t)b * K_ * K_);
    const uint32_t la = (uint32_t)(uintptr_t)(&Msh[0]);

    v4u g0;
    g0[0] = 1u;                                   // count=1, user descriptor
    g0[1] = la;                                   // lds_addr
    g0[2] = (uint32_t)(ga & 0xffffffffu);         // global_addr[31:0]
    g0[3] = (uint32_t)((ga >> 32) & 0x01ffffffu)  // global_addr[56:32]
            | (2u << 30);                         // type = 2 ("image")

    v8i g1;
    g1[0] = (int)((2u << 16)        // data_size = 4 bytes
                | (1u << 20)        // pad_enable
                | (6u << 22));      // pad_interval = 128 DWORDs, pad_amount=1dw
    g1[1] = (int)((uint32_t)K_ << 16);            // tensor_dim0[15:0] = 128
    g1[2] = (int)((uint32_t)K_ << 16);            // dim0[31:16]=0 | dim1[15:0]
    g1[3] = (int)((uint32_t)K_ << 16);            // dim1[31:16]=0 | tile_dim0
    g1[4] = (int)(uint32_t)K_;                    // tile_dim1=128, tile_dim2=0
    g1[5] = (int)(uint32_t)K_;                    // tensor_dim0_stride = 128
    g1[6] = (int)((uint32_t)(K_ * K_) << 16);     // dim0_stride hi=0 | dim1_stride lo
    g1[7] = (int)((uint32_t)(K_ * K_) >> 16);     // dim1_stride hi

    v4i gz4 = {};                                 // 2D tensor: groups 2/3 unused
    v8i gz8 = {};                                 // clang-23 6-arg form: extra group
    __builtin_amdgcn_tensor_load_to_lds(g0, g1, gz4, gz4, gz8, 0);
    __builtin_amdgcn_s_wait_tensorcnt((short)0);
  }

  // --- all threads: RHS into the pad column + MASK row term ---------------
  RHS(r) = AAyx[((size_t)b * K_ + i) * K_ + r];

  const float sc = scale[b];
  const float g1v = sqrtf(evx[b * K_ + r] / sc);  // gamma = 0.5 -> sqrt
  const float g2v = sqrtf(evy[b * K_ + i] / sc);
  const float d1 = 1.0f / (g1v * g1v + 1.0f);
  const float d2 = 1.0f / (g2v * g2v + 1.0f);
  const float re = g2v * d2 - g1v * d1;
  const float im = d2 - d1;
  const float mask = re * re + im * im;

  __syncthreads();                 // TDM tile + RHS visible to everyone
  MS(r, r) += LMBDA_ * mask;
  __syncthreads();

  // forward elimination (row r is eliminated against pivot rows j < r)
  for (int j = 0; j < K_; ++j) {
    if (r > j) {
      const float f = MS(r, j) / MS(j, j);
      for (int c = j + 1; c < K_; ++c) MS(r, c) -= f * MS(j, c);
      RHS(r) -= f * RHS(j);
    }
    __syncthreads();
  }

  // back substitution: x overwrites the RHS column in place
  for (int j = K_ - 1; j >= 0; --j) {
    if (r == j) RHS(j) = RHS(j) / MS(j, j);
    __syncthreads();
    if (r < j) RHS(r) -= MS(r, j) * RHS(j);
    __syncthreads();
  }

  out[((size_t)b * K_ + i) * K_ + r] = RHS(r);
}

// ---------------------------------------------------------------------------
// Host launcher
// inputs (setup_inputs order): evals_x, pinv_x, evals_y, pinv_y, feat_x, feat_y
// ---------------------------------------------------------------------------
extern "C" void kernel_launch(void* const* d_in, const int* in_sizes, int n_in,
                              void* d_out, int out_size, void* d_ws, size_t ws_size,
                              hipStream_t stream) {
  (void)in_sizes; (void)n_in; (void)out_size; (void)ws_size;

  const float* evx    = (const float*)d_in[0];
  const float* pinv_x = (const float*)d_in[1];
  const float* evy    = (const float*)d_in[2];
  const float* pinv_y = (const float*)d_in[3];
  const float* feat_x = (const float*)d_in[4];
  const float* feat_y = (const float*)d_in[5];

  // workspace layout (floats): Ax | Ay | AAxx | AAyx | scale   (~12.6 MB)
  float* Ax    = (float*)d_ws;
  float* Ay    = Ax + (size_t)B_ * K_ * C_;
  float* AAxx  = Ay + (size_t)B_ * K_ * C_;
  float* AAyx  = AAxx + (size_t)B_ * K_ * K_;
  float* scale = AAyx + (size_t)B_ * K_ * K_;

  fm_scale_kernel<<<dim3(B_), dim3(128), 0, stream>>>(evx, evy, scale);

  fm_proj_kernel<<<dim3(C_ / 16, K_ / 64, B_ * 2), dim3(32, 4), 0, stream>>>(
      pinv_x, feat_x, pinv_y, feat_y, Ax, Ay);

  fm_gram_kernel<<<dim3(K_ / 16, K_ / 16, B_ * 2), dim3(32), 0, stream>>>(
      Ax, Ay, AAxx, AAyx);

  fm_solve_kernel<<<dim3(K_, B_), dim3(128), 0, stream>>>(
      AAxx, AAyx, evx, evy, scale, (float*)d_out);
}